// div_minibatch_19026705121510
// MI455X (gfx1250) — compile-verified
//
#include <hip/hip_runtime.h>

// ---------------------------------------------------------------------------
// Depthwise periodic 3x3x3 stencil, fp32, CDNA5 (gfx1250).
// out[b,i,z,y,x] = sum_j sum_{dz,dy,dx} K[j,dz,dy,dx] *
//                  sig[b,i,j,(z+dz-1)%128,(y+dy-1)%128,(x+dx-1)%128]
// B=4, 3 components i, 3 kernels j, 128^3 volume.
// CDNA5 path: async global->LDS copies (ASYNCcnt) with a 4-slot rolling
// z-plane double buffer; compute is fp32 FMA out of LDS (ds_load_b128 +
// ds_load_2addr), weights as SGPR operands.
// (Packed V_PK_FMA_F32 variants were tried and regressed: +120 static VALU
//  from SGPR->VGPR pair copies and +80 VGPRs pressure; reverted.)
// ---------------------------------------------------------------------------

#define GLOBAL_AS __attribute__((address_space(1)))
#define LDS_AS    __attribute__((address_space(3)))

typedef int v4i __attribute__((vector_size(16)));

#if defined(__has_builtin)
#if __has_builtin(__builtin_amdgcn_global_load_async_to_lds_b128)
#define HAVE_ASYNC_LDS 1
#endif
#endif

__device__ __forceinline__ void copy16_g2lds(const float* __restrict__ g,
                                             float* __restrict__ l) {
#if defined(HAVE_ASYNC_LDS)
  __builtin_amdgcn_global_load_async_to_lds_b128(
      (GLOBAL_AS v4i*)g, (LDS_AS v4i*)l, /*offset=*/0, /*cpol=*/0);
#else
  *(float4*)l = *(const float4*)g;
#endif
}

__device__ __forceinline__ void wait_async_copies() {
#if defined(HAVE_ASYNC_LDS)
#if __has_builtin(__builtin_amdgcn_s_wait_asynccnt)
  __builtin_amdgcn_s_wait_asynccnt(0);
#else
  asm volatile("s_wait_asynccnt 0" ::: "memory");
#endif
#endif
}

__global__ __launch_bounds__(128) void stencil3x3x3_kernel(
    const float* __restrict__ sig,   // [4,3,3,128,128,128]
    const float* __restrict__ kern,  // [3,3,3,3]  (j,dz,dy,dx)
    float* __restrict__ out) {       // [4,3,128,128,128]
  constexpr int Wd  = 128;
  constexpr int HWd = 128 * 128;
  constexpr int CHd = 128 * 128 * 128;
  constexpr int PLANE = 30 * 128;      // 3 channels * 10 y-rows * 128 x
  __shared__ float lds[4 * PLANE];     // 4 rolling z-plane slots (60 KB)

  // Block decode: 4(b) * 3(i) * 16(y-tile) * 8(z-chunk) = 1536 blocks.
  int t  = (int)blockIdx.x;
  int zc = t & 7;  t >>= 3;
  int yt = t & 15; t >>= 4;
  int ci = t % 3;
  int b  = t / 3;
  int z0 = zc * 16;
  int y0 = yt * 8;

  int tid  = (int)threadIdx.x;
  int lane = tid & 31;   // x-group: this thread owns x0..x0+3
  int wv   = tid >> 5;   // wave id 0..3: owns output rows tyA, tyA+1
  int x0   = lane * 4;
  int tyA  = wv * 2;

  const float* gbi = sig + (size_t)((b * 3 + ci) * 3) * CHd;

  // Weights: contiguous scalar loads; uniform -> SGPR operands of the FMAs.
  float wk[81];
#pragma unroll
  for (int q = 0; q < 81; ++q) wk[q] = kern[q];

  // Load one z-plane (3 channels x 10 halo'd y-rows x 128 x) into slot (p+1)&3.
  auto loadPlane = [&](int p) {
    int zin  = (p + 128) & 127;
    int slot = (p + 1) & 3;          // p >= -1
    float* ls = lds + slot * PLANE;
#pragma unroll
    for (int k = 0; k < 8; ++k) {
      int r = wv + 4 * k;            // row 0..29 = j*10 + rr
      if (r < 30) {
        int j  = (r >= 20) ? 2 : (r >= 10 ? 1 : 0);
        int rr = r - j * 10;
        int yy = (y0 - 1 + rr + 128) & 127;
        const float* g = gbi + (size_t)j * CHd + (size_t)zin * HWd +
                         (size_t)yy * Wd + x0;
        copy16_g2lds(g, ls + r * 128 + x0);
      }
    }
  };

  // Prologue: planes z0-1, z0, z0+1.
  loadPlane(z0 - 1);
  loadPlane(z0);
  loadPlane(z0 + 1);
  wait_async_copies();
  __syncthreads();

  int xm1 = (x0 + 127) & 127;  // circular x halo indices (constant per thread)
  int xp4 = (x0 + 4) & 127;

  for (int z = z0; z < z0 + 16; ++z) {
    // Overlap: fetch plane z+2 into the free slot while computing z.
    if (z + 2 <= z0 + 16) loadPlane(z + 2);

    float acc[2][4];
#pragma unroll
    for (int o = 0; o < 2; ++o)
#pragma unroll
      for (int xi = 0; xi < 4; ++xi) acc[o][xi] = 0.0f;

#pragma unroll
    for (int j = 0; j < 3; ++j) {
#pragma unroll
      for (int dz = 0; dz < 3; ++dz) {
        int slot = (z + dz) & 3;     // plane p = z-1+dz -> slot (p+1)&3
        const float* base = lds + slot * PLANE + (j * 10 + tyA) * 128;
#pragma unroll
        for (int riy = 0; riy < 4; ++riy) {
          const float* row = base + riy * 128;
          float4 c = *(const float4*)(row + x0);   // ds_load_b128
          float vl = row[xm1];
          float vr = row[xp4];
          float v[6] = {vl, c.x, c.y, c.z, c.w, vr};
#pragma unroll
          for (int o = 0; o < 2; ++o) {
            int dy = riy - o;
            if (0 <= dy && dy < 3) {
              const float* wp = &wk[((j * 3 + dz) * 3 + dy) * 3];
#pragma unroll
              for (int dx = 0; dx < 3; ++dx)
#pragma unroll
                for (int xi = 0; xi < 4; ++xi)
                  acc[o][xi] = __builtin_fmaf(wp[dx], v[xi + dx], acc[o][xi]);
            }
          }
        }
      }
    }

    size_t ob = ((((size_t)(b * 3 + ci)) * 128 + z) * 128 + (y0 + tyA)) * 128 + x0;
    *(float4*)(out + ob)       = make_float4(acc[0][0], acc[0][1], acc[0][2], acc[0][3]);
    *(float4*)(out + ob + 128) = make_float4(acc[1][0], acc[1][1], acc[1][2], acc[1][3]);

    wait_async_copies();   // plane z+2 resident before next iteration reads it
    __syncthreads();
  }
}

extern "C" void kernel_launch(void* const* d_in, const int* in_sizes, int n_in,
                              void* d_out, int out_size, void* d_ws, size_t ws_size,
                              hipStream_t stream) {
  const float* sig  = (const float*)d_in[0];  // 4*9*128^3 fp32
  const float* kern = (const float*)d_in[1];  // 81 fp32
  float* out = (float*)d_out;                 // 4*3*128^3 fp32
  (void)in_sizes; (void)n_in; (void)out_size; (void)d_ws; (void)ws_size;

  dim3 grid(4 * 3 * 16 * 8);  // b * i * y-tiles * z-chunks = 1536
  dim3 block(128);
  stencil3x3x3_kernel<<<grid, block, 0, stream>>>(sig, kern, out);
}